// TorchVQCRouter_22617297780860
// MI455X (gfx1250) — compile-verified
//
#include <hip/hip_runtime.h>
#include <hip/hip_bf16.h>

typedef __attribute__((ext_vector_type(2))) float f32x2;
typedef __attribute__((ext_vector_type(8))) float v8f;

#define PI_F 3.14159265358979323846f

__device__ __forceinline__ float fast_tanh(float p) {
#if __has_builtin(__builtin_amdgcn_tanhf)
    return __builtin_amdgcn_tanhf(p);          // v_tanh_f32 (gfx1250 TRANS op)
#else
    // branch-free: tanh(p) = sign(p) * (1 - 2/(e^{2|p|}+1)); inf-safe.
    float a = __builtin_fabsf(p);
    float e = __expf(2.0f * a);
    float t = 1.0f - __fdividef(2.0f, e + 1.0f);
    return __builtin_copysignf(t, p);
#endif
}

__global__ __launch_bounds__(256)
void vqc_fused_kernel(const float* __restrict__ features,   // (B, 4096)
                      const float* __restrict__ proj_w,     // (16, 4096)
                      const float* __restrict__ proj_b,     // (16,)
                      const float* __restrict__ vqc_w,      // (24, 16, 3)
                      const float* __restrict__ lin_w,      // (16, 512)
                      const float* __restrict__ lin_b,      // (512,)
                      float* __restrict__ out)              // (B, 512)
{
    constexpr int D = 4096, Q = 16, L = 24, C = 512;
    constexpr int KHALF = D / 2;                 // K-split across wave pairs
    __shared__ float lds[4 * 16 * 18];           // 4 tiles/block, stride 18

    const int lane = threadIdx.x & 31;
    const int wave = threadIdx.x >> 5;
    const int lh   = lane >> 4;                  // lane K-half (A/B layout)
    const int m    = lane & 15;                  // A-row / B-col / C-col
    const int pair = wave >> 1;                  // tile index within block
    const int ks   = wave & 1;                   // K-split half of this wave
    const int tile = blockIdx.x * 4 + pair;
    const int row0 = tile * 16;

    // Last tile's odd wave would prefetch past the end of `features`;
    // WGP-scope prefetch is non-speculative, so skip it there. The condition
    // is wave-uniform by construction; readfirstlane makes that visible to
    // the compiler so the guard is a scalar branch (no EXEC manipulation in
    // the WMMA loop).
    const int pf_ok = __builtin_amdgcn_readfirstlane(
        ((tile != (int)(gridDim.x * 4 - 1)) || (ks == 0)) ? 1 : 0);

    // ---------------- Stage 1: partial proj over this wave's K half --------
    const float* arow = features + (size_t)(row0 + m) * D + ks * KHALF + 2 * lh;
    const float* brow = proj_w   + (size_t)m        * D + ks * KHALF + 2 * lh;
    const f32x2* ap = reinterpret_cast<const f32x2*>(arow);
    const f32x2* bp = reinterpret_cast<const f32x2*>(brow);

    v8f acc[4] = {v8f{}, v8f{}, v8f{}, v8f{}};
#pragma unroll 1
    for (int k = 0; k < KHALF; k += 64) {        // 64 K per outer iteration
        // near-scope prefetch of the A stream ~4KB ahead (2 lines per row)
        if (pf_ok) {
            __builtin_prefetch(arow + k + 1024, 0, 3);
            __builtin_prefetch(arow + k + 1056, 0, 3);
        }
#pragma unroll
        for (int s = 0; s < 8; ++s) {
            const int kk = k + s * 8;
            f32x2 a0 = ap[(kk >> 1) + 0];
            f32x2 b0 = bp[(kk >> 1) + 0];
            f32x2 a1 = ap[(kk >> 1) + 2];
            f32x2 b1 = bp[(kk >> 1) + 2];
            acc[(2 * s) & 3] = __builtin_amdgcn_wmma_f32_16x16x4_f32(
                false, a0, false, b0, (short)0, acc[(2 * s) & 3], false, false);
            acc[(2 * s + 1) & 3] = __builtin_amdgcn_wmma_f32_16x16x4_f32(
                false, a1, false, b1, (short)0, acc[(2 * s + 1) & 3], false, false);
        }
    }

    float psum[8];
#pragma unroll
    for (int r = 0; r < 8; ++r)
        psum[r] = (acc[0][r] + acc[1][r]) + (acc[2][r] + acc[3][r]);

    // ------- pair reduction: even wave publishes its partial tile ----------
    float* tl = &lds[pair * (16 * 18)];
    if (ks == 0) {
#pragma unroll
        for (int r = 0; r < 8; ++r)
            tl[(r + 8 * lh) * 18 + m] = psum[r];
    }
    __syncthreads();

    // ------- odd wave: reduce, bias, tanh, VQC; publish z tile -------------
    if (ks == 1) {
        const float pb = proj_b[m];
        float xv[8], yv[8], zv[8];
#pragma unroll
        for (int r = 0; r < 8; ++r) {
            float p   = psum[r] + tl[(r + 8 * lh) * 18 + m] + pb;
            float ang = PI_F * fast_tanh(p);
            __sincosf(ang, &xv[r], &zv[r]);      // x = sin, z = cos
            yv[r] = 0.0f;
        }
#pragma unroll 1
        for (int l = 0; l < L; ++l) {
            const float* w = vqc_w + (l * Q + m) * 3;
            float sp, cp, st, ct, so, co;
            __sincosf(w[0], &sp, &cp);
            __sincosf(w[1], &st, &ct);
            __sincosf(w[2], &so, &co);
#pragma unroll
            for (int r = 0; r < 8; ++r) {
                float x1 = xv[r] * cp - yv[r] * sp;
                float y1 = xv[r] * sp + yv[r] * cp;
                float x2 = x1 * ct + zv[r] * st;
                float z2 = -x1 * st + zv[r] * ct;
                xv[r] = x2 * co - y1 * so;
                yv[r] = x2 * so + y1 * co;
                zv[r] = z2;
            }
        }
#pragma unroll
        for (int r = 0; r < 8; ++r)
            tl[(r + 8 * lh) * 18 + m] = zv[r];   // overwrite tile with z
    }
    __syncthreads();

    // ------- Stage 3: both waves of the pair split the N range -------------
    f32x2 a2[4];
#pragma unroll
    for (int kk = 0; kk < 4; ++kk)
        a2[kk] = *reinterpret_cast<const f32x2*>(&tl[m * 18 + kk * 4 + 2 * lh]);

    const float* lwbase = lin_w + (size_t)(2 * lh) * C + m;
    const int nbeg = ks * (C / 2);
#pragma unroll 1
    for (int n0 = nbeg; n0 < nbeg + C / 2; n0 += 16) {
        v8f c = {};
#pragma unroll
        for (int kk = 0; kk < 4; ++kk) {
            f32x2 b;
            b.x = lwbase[(size_t)(kk * 4) * C + n0];
            b.y = lwbase[(size_t)(kk * 4 + 1) * C + n0];
            c = __builtin_amdgcn_wmma_f32_16x16x4_f32(
                    false, a2[kk], false, b, (short)0, c, false, false);
        }
        const float bias = lin_b[n0 + m];
        float* orow = out + (size_t)(row0 + 8 * lh) * C + n0 + m;
#pragma unroll
        for (int r = 0; r < 8; ++r)
            __builtin_nontemporal_store(c[r] + bias, orow + (size_t)r * C);
    }
}

extern "C" void kernel_launch(void* const* d_in, const int* in_sizes, int n_in,
                              void* d_out, int out_size, void* d_ws, size_t ws_size,
                              hipStream_t stream) {
    const float* features = (const float*)d_in[0];
    const float* proj_w   = (const float*)d_in[1];
    const float* proj_b   = (const float*)d_in[2];
    const float* vqc_w    = (const float*)d_in[3];
    const float* lin_w    = (const float*)d_in[4];
    const float* lin_b    = (const float*)d_in[5];
    float* out = (float*)d_out;

    const int B = in_sizes[0] / 4096;   // 8192 rows
    const int blocks = B / 64;          // 4 tiles/block * 16 rows, 2 waves/tile
    vqc_fused_kernel<<<blocks, 256, 0, stream>>>(
        features, proj_w, proj_b, vqc_w, lin_w, lin_b, out);
}